// GNNVirtual_node_prop_39616778338396
// MI455X (gfx1250) — compile-verified
//
#include <hip/hip_runtime.h>
#include <hip/hip_bf16.h>

typedef float v2f __attribute__((ext_vector_type(2)));
typedef float v8f __attribute__((ext_vector_type(8)));
typedef int v4i __attribute__((ext_vector_type(4)));

#define D128 128

// ---- gfx1250 async global->LDS (signature learned from compiler diagnostic) ----
#if defined(__HIP_DEVICE_COMPILE__) && defined(__gfx1250__) && \
    __has_builtin(__builtin_amdgcn_global_load_async_to_lds_b128)
#define USE_ASYNC_LDS 1
typedef __attribute__((address_space(1))) v4i* as1v4p;
typedef __attribute__((address_space(3))) v4i* as3v4p;
__device__ __forceinline__ void asyncCopy16(const float* g, float* l) {
    __builtin_amdgcn_global_load_async_to_lds_b128((as1v4p)(v4i*)g, (as3v4p)(v4i*)l,
                                                   0, 0);
}
__device__ __forceinline__ void asyncWait0() {
#if __has_builtin(__builtin_amdgcn_s_wait_asynccnt)
    __builtin_amdgcn_s_wait_asynccnt(0);
#else
    asm volatile("s_wait_asynccnt 0x0" ::: "memory");
#endif
}
#else
#define USE_ASYNC_LDS 0
#endif

__device__ __forceinline__ void atomAddF(float* p, float v) {
#if defined(__gfx1250__)
    unsafeAtomicAdd(p, v);
#else
    atomicAdd(p, v);
#endif
}

// ---------------- utility kernels ----------------

__global__ void zerok(float* __restrict__ p, size_t n) {
    size_t t = (size_t)blockIdx.x * blockDim.x + threadIdx.x;
    size_t stride = (size_t)gridDim.x * blockDim.x;
    for (; t < n; t += stride) p[t] = 0.0f;
}

__global__ void hinit(float* __restrict__ h, const int* __restrict__ x,
                      const int* __restrict__ nd,
                      const float* __restrict__ te, const float* __restrict__ ae,
                      const float* __restrict__ de, int n) {
    int t = blockIdx.x * blockDim.x + threadIdx.x;
    if (t >= n) return;
    int i = t >> 7, c = t & 127;
    h[t] = te[(size_t)x[2 * i] * D128 + c] +
           ae[(size_t)x[2 * i + 1] * D128 + c] +
           de[(size_t)nd[i] * D128 + c];
}

__global__ void vninit(float* __restrict__ vn, const float* __restrict__ vne) {
    int t = blockIdx.x * blockDim.x + threadIdx.x;
    if (t < 128 * D128) vn[t] = vne[t & 127];
}

__global__ void addvn(float* __restrict__ h, const float* __restrict__ vn,
                      const int* __restrict__ batch, int n, int dopool,
                      float* __restrict__ vnpool) {
    int t = blockIdx.x * blockDim.x + threadIdx.x;
    if (t >= n) return;
    int i = t >> 7, c = t & 127;
    int g = batch[i];
    float hv = h[t] + vn[g * D128 + c];
    h[t] = hv;
    if (dopool) atomAddF(&vnpool[g * D128 + c], hv);
}

// one wave per edge, each lane handles 4 columns:
// msg = relu(h[src] + ea0*W[0,:] + ea1*W[1,:] + eb); atomic scatter into aggr[dst]
__global__ void edge_msg(const int* __restrict__ ei, const float* __restrict__ ea,
                         const float* __restrict__ W, const float* __restrict__ eb,
                         const float* __restrict__ h, float* __restrict__ aggr, int E_) {
    int t = blockIdx.x * blockDim.x + threadIdx.x;
    int e = t >> 5;
    if (e >= E_) return;
    int lane = t & 31;
    int c = lane * 4;
    int src = ei[e], dst = ei[E_ + e];
    float a0 = ea[2 * e], a1 = ea[2 * e + 1];
    float4 w0 = *(const float4*)(W + c);
    float4 w1 = *(const float4*)(W + D128 + c);
    float4 bb = *(const float4*)(eb + c);
    float4 hh = *(const float4*)(h + (size_t)src * D128 + c);
    float m0 = fmaxf(hh.x + a0 * w0.x + a1 * w1.x + bb.x, 0.0f);
    float m1 = fmaxf(hh.y + a0 * w0.y + a1 * w1.y + bb.y, 0.0f);
    float m2 = fmaxf(hh.z + a0 * w0.z + a1 * w1.z + bb.z, 0.0f);
    float m3 = fmaxf(hh.w + a0 * w0.w + a1 * w1.w + bb.w, 0.0f);
    float* ap = aggr + (size_t)dst * D128 + c;
    atomAddF(ap + 0, m0);
    atomAddF(ap + 1, m1);
    atomAddF(ap + 2, m2);
    atomAddF(ap + 3, m3);
}

// -------- WMMA GEMMs: 256-thread block = 8 waves; LDS-staged 16-row A strip --------
// LDS row stride padded (132 / 264 floats) so the 16 column-strided fragment reads
// hit distinct banks (128 % 64 == 0 would be a 16-way conflict).

// y[N,256] = ((1+eps)*h + aggr)[N,128] @ W1[128,256] + b1
// grid: (ceil(N/16), 2); wave w handles col tile blockIdx.y*8 + w
__global__ __launch_bounds__(256) void gemm1(const float* __restrict__ h,
                                             const float* __restrict__ aggr,
                                             const float* __restrict__ W,
                                             const float* __restrict__ bias,
                                             const float* __restrict__ epsArr, int l,
                                             float* __restrict__ y, int Nrows) {
    const int C = 256, LDW = 132;
    __shared__ float lds_h[16 * LDW];
    __shared__ float lds_g[16 * LDW];
    int t = threadIdx.x;
    int row0 = blockIdx.x * 16;

    // stage raw h and aggr tiles (16 x 128); OOB rows clamped to row 0
    {
        int sr = t >> 4;            // 0..15
        int sc = (t & 15) * 8;      // 0,8,...,120
        int grow = row0 + sr;
        const float* hp = h + (size_t)((grow < Nrows) ? grow : 0) * D128 + sc;
        const float* gp = aggr + (size_t)((grow < Nrows) ? grow : 0) * D128 + sc;
        float* dh = &lds_h[sr * LDW + sc];
        float* dg = &lds_g[sr * LDW + sc];
#if USE_ASYNC_LDS
        asyncCopy16(hp, dh);
        asyncCopy16(hp + 4, dh + 4);
        asyncCopy16(gp, dg);
        asyncCopy16(gp + 4, dg + 4);
        asyncWait0();
#else
        *(float4*)dh = *(const float4*)hp;
        *(float4*)(dh + 4) = *(const float4*)(hp + 4);
        *(float4*)dg = *(const float4*)gp;
        *(float4*)(dg + 4) = *(const float4*)(gp + 4);
#endif
    }
    __syncthreads();

    int lane = t & 31;
    int w = t >> 5;  // 0..7
    int m = lane & 15, kh = lane >> 4;
    bool rowok = (row0 + m) < Nrows;
    int n = (blockIdx.y * 8 + w) * 16 + (lane & 15);
    float eps1 = 1.0f + epsArr[l];

    v8f acc = {};
    for (int kb = 0; kb < 128; kb += 4) {
        int k0 = kb + 2 * kh;
        v2f a, b;
        float z0 = eps1 * lds_h[m * LDW + k0] + lds_g[m * LDW + k0];
        float z1 = eps1 * lds_h[m * LDW + k0 + 1] + lds_g[m * LDW + k0 + 1];
        a.x = rowok ? z0 : 0.0f;
        a.y = rowok ? z1 : 0.0f;
        b.x = W[(size_t)k0 * C + n];
        b.y = W[(size_t)(k0 + 1) * C + n];
        acc = __builtin_amdgcn_wmma_f32_16x16x4_f32(false, a, false, b, (short)0, acc,
                                                    false, false);
    }
    float bv = bias[n];
#pragma unroll
    for (int j = 0; j < 8; j++) {
        int r = row0 + j + 8 * kh;
        if (r < Nrows) y[(size_t)r * C + n] = acc[j] + bv;
    }
}

// z2[N,128] = relu(y*a1+c1)[N,256] @ W2[256,128] + b2
// grid: (ceil(N/16), 1); wave w handles col tile w
__global__ __launch_bounds__(256) void gemm2(const float* __restrict__ y,
                                             const float* __restrict__ a1,
                                             const float* __restrict__ c1,
                                             const float* __restrict__ W,
                                             const float* __restrict__ bias,
                                             float* __restrict__ z2, int Nrows) {
    const int Cin = 256, Cout = 128, LDW = 264;
    __shared__ float lds_y[16 * LDW];
    int t = threadIdx.x;
    int row0 = blockIdx.x * 16;

    {
        int sr = t >> 4;             // 0..15
        int sc = (t & 15) * 16;      // 0,16,...,240
        int grow = row0 + sr;
        const float* yp = y + (size_t)((grow < Nrows) ? grow : 0) * Cin + sc;
        float* dy = &lds_y[sr * LDW + sc];
#if USE_ASYNC_LDS
        asyncCopy16(yp, dy);
        asyncCopy16(yp + 4, dy + 4);
        asyncCopy16(yp + 8, dy + 8);
        asyncCopy16(yp + 12, dy + 12);
        asyncWait0();
#else
        *(float4*)dy = *(const float4*)yp;
        *(float4*)(dy + 4) = *(const float4*)(yp + 4);
        *(float4*)(dy + 8) = *(const float4*)(yp + 8);
        *(float4*)(dy + 12) = *(const float4*)(yp + 12);
#endif
    }
    __syncthreads();

    int lane = t & 31;
    int w = t >> 5;
    int m = lane & 15, kh = lane >> 4;
    bool rowok = (row0 + m) < Nrows;
    int n = w * 16 + (lane & 15);

    v8f acc = {};
    for (int kb = 0; kb < 256; kb += 4) {
        int k0 = kb + 2 * kh;
        v2f a, b;
        float x0 = fmaxf(lds_y[m * LDW + k0] * a1[k0] + c1[k0], 0.0f);
        float x1 = fmaxf(lds_y[m * LDW + k0 + 1] * a1[k0 + 1] + c1[k0 + 1], 0.0f);
        a.x = rowok ? x0 : 0.0f;
        a.y = rowok ? x1 : 0.0f;
        b.x = W[(size_t)k0 * Cout + n];
        b.y = W[(size_t)(k0 + 1) * Cout + n];
        acc = __builtin_amdgcn_wmma_f32_16x16x4_f32(false, a, false, b, (short)0, acc,
                                                    false, false);
    }
    float bv = bias[n];
#pragma unroll
    for (int j = 0; j < 8; j++) {
        int r = row0 + j + 8 * kh;
        if (r < Nrows) z2[(size_t)r * Cout + n] = acc[j] + bv;
    }
}

// ---------------- BatchNorm support ----------------

__global__ void colstats(const float* __restrict__ M, int Nrows, int C,
                         float* __restrict__ sum, float* __restrict__ sq) {
    int c = threadIdx.x;
    int chunk = (Nrows + gridDim.x - 1) / gridDim.x;
    int r0 = blockIdx.x * chunk;
    int r1 = r0 + chunk;
    if (r1 > Nrows) r1 = Nrows;
    float s = 0.0f, q = 0.0f;
    for (int r = r0; r < r1; r++) {
        float v = M[(size_t)r * C + c];
        s += v;
        q += v * v;
    }
    atomAddF(&sum[c], s);
    atomAddF(&sq[c], q);
}

__global__ void bnfin(const float* __restrict__ sum, const float* __restrict__ sq,
                      const float* __restrict__ g, const float* __restrict__ b,
                      float inv_n, float* __restrict__ a, float* __restrict__ c) {
    int i = threadIdx.x;
    float mu = sum[i] * inv_n;
    float var = sq[i] * inv_n - mu * mu;
    float inv = rsqrtf(var + 1e-5f);
    float ai = g[i] * inv;
    a[i] = ai;
    c[i] = b[i] - mu * ai;
}

__global__ void applybn(const float* __restrict__ z, const float* __restrict__ a,
                        const float* __restrict__ c, int dorelu,
                        float* __restrict__ out, int n) {
    int t = blockIdx.x * blockDim.x + threadIdx.x;
    if (t >= n) return;
    int col = t & 127;
    float v = z[t] * a[col] + c[col];
    if (dorelu) v = fmaxf(v, 0.0f);
    out[t] = v;
}

// ---------------- virtual-node MLP (tiny: G=128 rows), single block ----------------

__global__ __launch_bounds__(256) void vn_update(
    const float* __restrict__ vnpool, float* __restrict__ vn,
    const float* __restrict__ W1, const float* __restrict__ b1,
    const float* __restrict__ g1, const float* __restrict__ bb1,
    const float* __restrict__ W2, const float* __restrict__ b2,
    const float* __restrict__ g2, const float* __restrict__ bb2,
    float* __restrict__ vt, float* __restrict__ v2b) {
    int c = threadIdx.x;
    float s = 0.0f, q = 0.0f;
    for (int g = 0; g < 128; g++) {
        const float* xp = vnpool + (size_t)g * D128;
        const float* xv = vn + (size_t)g * D128;
        float acc = b1[c];
        for (int k = 0; k < 128; k++) acc += (xp[k] + xv[k]) * W1[(size_t)k * 256 + c];
        vt[(size_t)g * 256 + c] = acc;
        s += acc;
        q += acc * acc;
    }
    {
        float mu = s * (1.0f / 128.0f);
        float var = q * (1.0f / 128.0f) - mu * mu;
        float inv = rsqrtf(var + 1e-5f);
        float aa = g1[c] * inv, cc = bb1[c] - mu * aa;
        for (int g = 0; g < 128; g++) {
            float v = vt[(size_t)g * 256 + c];
            vt[(size_t)g * 256 + c] = fmaxf(v * aa + cc, 0.0f);
        }
    }
    __threadfence();
    __syncthreads();
    if (c < 128) {
        float s2 = 0.0f, q2 = 0.0f;
        for (int g = 0; g < 128; g++) {
            const float* xr = vt + (size_t)g * 256;
            float acc = b2[c];
            for (int k = 0; k < 256; k++) acc += xr[k] * W2[(size_t)k * D128 + c];
            v2b[(size_t)g * D128 + c] = acc;
            s2 += acc;
            q2 += acc * acc;
        }
        float mu = s2 * (1.0f / 128.0f);
        float var = q2 * (1.0f / 128.0f) - mu * mu;
        float inv = rsqrtf(var + 1e-5f);
        float aa = g2[c] * inv, cc = bb2[c] - mu * aa;
        for (int g = 0; g < 128; g++)
            vn[(size_t)g * D128 + c] = fmaxf(v2b[(size_t)g * D128 + c] * aa + cc, 0.0f);
    }
}

// ---------------- driver ----------------

extern "C" void kernel_launch(void* const* d_in, const int* in_sizes, int n_in,
                              void* d_out, int out_size, void* d_ws, size_t ws_size,
                              hipStream_t stream) {
    const int* x = (const int*)d_in[0];
    const int* node_depth = (const int*)d_in[1];
    const int* edge_index = (const int*)d_in[2];
    const int* batch = (const int*)d_in[3];
    const float* edge_attr = (const float*)d_in[4];
    const float* type_emb = (const float*)d_in[5];
    const float* attr_emb = (const float*)d_in[6];
    const float* depth_emb = (const float*)d_in[7];
    const float* vn_emb = (const float*)d_in[8];
    const float* epsArr = (const float*)d_in[9];
    const float* edge_W = (const float*)d_in[10];
    const float* edge_b = (const float*)d_in[11];
    const float* gin_W1 = (const float*)d_in[12];
    const float* gin_b1 = (const float*)d_in[13];
    const float* gin_bng = (const float*)d_in[14];
    const float* gin_bnb = (const float*)d_in[15];
    const float* gin_W2 = (const float*)d_in[16];
    const float* gin_b2 = (const float*)d_in[17];
    const float* bng = (const float*)d_in[18];
    const float* bnb = (const float*)d_in[19];
    const float* vn_W1 = (const float*)d_in[20];
    const float* vn_b1 = (const float*)d_in[21];
    const float* vn_bn1g = (const float*)d_in[22];
    const float* vn_bn1b = (const float*)d_in[23];
    const float* vn_W2 = (const float*)d_in[24];
    const float* vn_b2 = (const float*)d_in[25];
    const float* vn_bn2g = (const float*)d_in[26];
    const float* vn_bn2b = (const float*)d_in[27];

    const int N_ = in_sizes[1];
    const int E_ = in_sizes[4] / 2;
    const int L_ = 5, G_ = 128;

    float* ws = (float*)d_ws;
    size_t off = 0;
    float* h = ws + off;      off += (size_t)N_ * 128;
    float* aggr = ws + off;   off += (size_t)N_ * 128;
    float* y = ws + off;      off += (size_t)N_ * 256;
    float* z2 = ws + off;     off += (size_t)N_ * 128;
    float* vn = ws + off;     off += (size_t)G_ * 128;
    float* vnpool = ws + off; off += (size_t)G_ * 128;
    float* vt = ws + off;     off += (size_t)G_ * 256;
    float* v2b = ws + off;    off += (size_t)G_ * 128;
    float* stats = ws + off;  off += 4096;
    float* s1 = stats, *q1 = stats + 256, *a1 = stats + 512, *c1 = stats + 768;
    float* s2 = stats + 1024, *q2 = stats + 1152, *a2 = stats + 1280, *c2 = stats + 1408;

    const int nh = N_ * 128;
    const int rowTiles = (N_ + 15) / 16;

    hinit<<<(nh + 255) / 256, 256, 0, stream>>>(h, x, node_depth, type_emb, attr_emb,
                                                depth_emb, nh);
    vninit<<<64, 256, 0, stream>>>(vn, vn_emb);

    for (int l = 0; l < L_; l++) {
        zerok<<<2048, 256, 0, stream>>>(aggr, (size_t)N_ * 128);
        zerok<<<64, 256, 0, stream>>>(vnpool, (size_t)G_ * 128);
        zerok<<<8, 256, 0, stream>>>(stats, 2048);

        addvn<<<(nh + 255) / 256, 256, 0, stream>>>(h, vn, batch, nh,
                                                    (l < L_ - 1) ? 1 : 0, vnpool);

        edge_msg<<<(E_ * 32 + 255) / 256, 256, 0, stream>>>(
            edge_index, edge_attr, edge_W + (size_t)l * 2 * 128,
            edge_b + (size_t)l * 128, h, aggr, E_);

        gemm1<<<dim3(rowTiles, 2), 256, 0, stream>>>(
            h, aggr, gin_W1 + (size_t)l * 128 * 256, gin_b1 + (size_t)l * 256, epsArr,
            l, y, N_);
        colstats<<<120, 256, 0, stream>>>(y, N_, 256, s1, q1);
        bnfin<<<1, 256, 0, stream>>>(s1, q1, gin_bng + (size_t)l * 256,
                                     gin_bnb + (size_t)l * 256, 1.0f / (float)N_, a1, c1);

        gemm2<<<dim3(rowTiles, 1), 256, 0, stream>>>(
            y, a1, c1, gin_W2 + (size_t)l * 256 * 128, gin_b2 + (size_t)l * 128, z2, N_);
        colstats<<<120, 128, 0, stream>>>(z2, N_, 128, s2, q2);
        bnfin<<<1, 128, 0, stream>>>(s2, q2, bng + (size_t)l * 128,
                                     bnb + (size_t)l * 128, 1.0f / (float)N_, a2, c2);

        float* dst = (l == L_ - 1) ? (float*)d_out : h;
        applybn<<<(nh + 255) / 256, 256, 0, stream>>>(z2, a2, c2,
                                                      (l < L_ - 1) ? 1 : 0, dst, nh);

        if (l < L_ - 1) {
            vn_update<<<1, 256, 0, stream>>>(
                vnpool, vn, vn_W1 + (size_t)l * 128 * 256, vn_b1 + (size_t)l * 256,
                vn_bn1g + (size_t)l * 256, vn_bn1b + (size_t)l * 256,
                vn_W2 + (size_t)l * 256 * 128, vn_b2 + (size_t)l * 128,
                vn_bn2g + (size_t)l * 128, vn_bn2b + (size_t)l * 128, vt, v2b);
        }
    }
}